// GPT2StaticKVCache_46119358824834
// MI455X (gfx1250) — compile-verified
//
#include <hip/hip_runtime.h>
#include <hip/hip_bf16.h>
#include <math.h>

// Problem constants (GPT-2 decode step)
#define Lyr 12
#define Bsz 8
#define Hds 12
#define Emb 768
#define Dh  64
#define Seq 1024
#define Voc 50257
#define BHSD ((size_t)Bsz * Hds * Seq * Dh)   // per-layer cache elems: 6,291,456

// Split-K GEMM tiling
#define KCH 192                   // K-chunk per block (768=4*192, 3072=16*192)
#define SXP (KCH + 4)             // LDS row stride: 196 % 64banks = 4 -> conflict-free

typedef __attribute__((ext_vector_type(2))) float v2f;
typedef __attribute__((ext_vector_type(4))) float v4f;
typedef __attribute__((ext_vector_type(8))) float v8f;

__device__ __forceinline__ float gelu_tanh(float x) {
    const float c = 0.7978845608028654f; // sqrt(2/pi)
    float x3 = x * x * x;
    return 0.5f * x * (1.0f + tanhf(c * (x + 0.044715f * x3)));
}

// ---------------------------------------------------------------------------
// h[b][e] = wte[ids[b]][e] + wpe[pos][e]
// ---------------------------------------------------------------------------
__global__ void embed_kernel(const int* __restrict__ ids, const int* __restrict__ posp,
                             const float* __restrict__ wte, const float* __restrict__ wpe,
                             float* __restrict__ h) {
    int i = blockIdx.x * blockDim.x + threadIdx.x;
    if (i >= Bsz * Emb) return;
    int b = i / Emb, e = i - b * Emb;
    h[i] = wte[(size_t)ids[b] * Emb + e] + wpe[(size_t)(*posp) * Emb + e];
}

// ---------------------------------------------------------------------------
// Row-wise LayerNorm: Y[row] = (X[row]-mean)*rsqrt(var+eps)*g + b   (8 rows)
// ---------------------------------------------------------------------------
__global__ void layernorm_kernel(const float* __restrict__ X,
                                 const float* __restrict__ g, const float* __restrict__ bb,
                                 float* __restrict__ Y) {
    __shared__ float rs[256], rs2[256];
    int row = blockIdx.x;
    int tid = threadIdx.x;
    const float* xr = X + (size_t)row * Emb;
    float s = 0.f, s2 = 0.f;
    for (int e = tid; e < Emb; e += 256) { float v = xr[e]; s += v; s2 += v * v; }
    rs[tid] = s; rs2[tid] = s2;
    __syncthreads();
    for (int off = 128; off > 0; off >>= 1) {
        if (tid < off) { rs[tid] += rs[tid + off]; rs2[tid] += rs2[tid + off]; }
        __syncthreads();
    }
    float mean = rs[0] * (1.0f / Emb);
    float var  = rs2[0] * (1.0f / Emb) - mean * mean;
    float rstd = rsqrtf(var + 1e-5f);
    for (int e = tid; e < Emb; e += 256)
        Y[(size_t)row * Emb + e] = (xr[e] - mean) * rstd * g[e] + bb[e];
}

// ---------------------------------------------------------------------------
// Split-K WMMA GEMM partial: parts[sk][8 x N] = X[8 x Kchunk] @ W[Kchunk x N]
// grid = (N/128, K/KCH); 8 waves/block, one 16-col tile per wave.
// X chunk staged in LDS as 16 rows (rows 8..15 zero) -> branchless A fragments.
// W streamed nontemporally, double-buffered, 32-bit offsets from uniform base
// (SADDR + voffset form -> 32-bit address increments in the hot loop).
// A tile (16x4 f32): lanes 0-15 -> M=lane, V0=K0,V1=K1; lanes 16-31 -> K2/K3.
// B tile (4x16 f32): lanes 0-15 -> N=lane, V0=K0,V1=K1; lanes 16-31 -> K2/K3.
// D (16x16 f32): VGPR r, lanes 0-15 -> M=r, N=lane  => rows 0..7 stored.
// ---------------------------------------------------------------------------
__global__ void gemm_splitk_wmma(const float* __restrict__ X, const float* __restrict__ W,
                                 float* __restrict__ parts, int K, int N) {
    __shared__ float sx[16 * SXP];
    int tid  = threadIdx.x;
    int lane = tid & 31;
    int wave = tid >> 5;
    int k0   = blockIdx.y * KCH;

    // Stage X chunk: rows 0..7 real, rows 8..15 zero (batch padding).
    for (int i = tid; i < 8 * KCH; i += 256) {
        int r = i / KCH, c = i - r * KCH;
        sx[r * SXP + c] = X[(size_t)r * K + k0 + c];
    }
    for (int i = tid; i < 8 * KCH; i += 256) {
        int r = i / KCH, c = i - r * KCH;
        sx[(r + 8) * SXP + c] = 0.f;
    }
    __syncthreads();

    int tile = blockIdx.x * 8 + wave;
    int col0 = tile * 16;
    if (col0 >= N) return;
    int col  = col0 + (lane & 15);
    int koff = (lane >> 4) * 2;          // 0 (lanes 0-15) or 2 (lanes 16-31)
    int arow = lane & 15;

    // 32-bit element offsets from uniform base W (weights < 2^31 elements).
    unsigned uN   = (unsigned)N;
    unsigned off  = (unsigned)(k0 + koff) * uN + (unsigned)col;
    unsigned step = 4u * uN;

    v2f bc;
    bc.x = __builtin_nontemporal_load(W + off);
    bc.y = __builtin_nontemporal_load(W + off + uN);

    v8f acc = {};
    #pragma unroll 4
    for (int kc = 0; kc < KCH - 4; kc += 4) {
        unsigned offn = off + step;
        v2f bn;
        bn.x = __builtin_nontemporal_load(W + offn);
        bn.y = __builtin_nontemporal_load(W + offn + uN);
        __builtin_prefetch(W + off + 16u * uN, 0, 3);   // near-cache stream-ahead
        v2f a;
        a.x = sx[arow * SXP + kc + koff];
        a.y = sx[arow * SXP + kc + koff + 1];
        acc = __builtin_amdgcn_wmma_f32_16x16x4_f32(false, a, false, bc,
                                                    (short)0, acc, false, false);
        bc  = bn;
        off = offn;
    }
    {   // last K-step
        v2f a;
        a.x = sx[arow * SXP + (KCH - 4) + koff];
        a.y = sx[arow * SXP + (KCH - 4) + koff + 1];
        acc = __builtin_amdgcn_wmma_f32_16x16x4_f32(false, a, false, bc,
                                                    (short)0, acc, false, false);
    }

    if (lane < 16) {
        float* Pp = parts + (size_t)blockIdx.y * (Bsz * (size_t)N);
        #pragma unroll
        for (int r = 0; r < Bsz; ++r)
            Pp[(size_t)r * N + col] = acc[r];
    }
}

// ---------------------------------------------------------------------------
// Epilogue: Y[i] = act( bias[i%N] + sum_p parts[p][i] ) (+ residual[i])
// act: 0 = none, 1 = gelu
// ---------------------------------------------------------------------------
__global__ void reduce_epilogue_kernel(const float* __restrict__ parts,
                                       const float* __restrict__ bias,
                                       const float* __restrict__ residual,
                                       float* __restrict__ Y,
                                       int n, int N, int nparts, int act) {
    int i = blockIdx.x * blockDim.x + threadIdx.x;
    if (i >= n) return;
    int c = i - (i / N) * N;
    float s = bias[c];
    for (int p = 0; p < nparts; ++p)
        s += parts[(size_t)p * n + i];
    if (act == 1) s = gelu_tanh(s);
    if (residual) s += residual[i];
    Y[i] = s;
}

// ---------------------------------------------------------------------------
// Scatter new K/V rows (from qkv) into the output caches at position pos.
// ---------------------------------------------------------------------------
__global__ void scatter_kv_kernel(const float* __restrict__ qkv,
                                  float* __restrict__ nk, float* __restrict__ nv,
                                  const int* __restrict__ posp) {
    int i = blockIdx.x * blockDim.x + threadIdx.x;   // B*H*D = 6144
    if (i >= Bsz * Hds * Dh) return;
    int b = i / (Hds * Dh);
    int r = i - b * (Hds * Dh);                      // h*64 + d
    int pos = *posp;
    int hh = r / Dh, d = r - hh * Dh;
    size_t dst = (((size_t)b * Hds + hh) * Seq + pos) * Dh + d;
    nk[dst] = qkv[(size_t)b * (3 * Emb) + Emb     + r];
    nv[dst] = qkv[(size_t)b * (3 * Emb) + 2 * Emb + r];
}

// ---------------------------------------------------------------------------
// Attention for one (b,h) per block: scores over kv=pos+1, softmax, weighted V.
// ---------------------------------------------------------------------------
__global__ void attention_kernel(const float* __restrict__ qkv,
                                 const float* __restrict__ Kc, const float* __restrict__ Vc,
                                 const int* __restrict__ posp, float* __restrict__ O) {
    __shared__ float q[Dh];
    __shared__ float sc[Seq];
    __shared__ float red[256];
    int b  = blockIdx.x / Hds;
    int hh = blockIdx.x - b * Hds;
    int tid = threadIdx.x;
    int kv = *posp + 1;
    const float scale = 0.125f;      // 1/sqrt(64)

    if (tid < Dh) q[tid] = qkv[(size_t)b * (3 * Emb) + hh * Dh + tid];
    __syncthreads();

    const float* Kb = Kc + ((size_t)b * Hds + hh) * Seq * Dh;
    const float* Vb = Vc + ((size_t)b * Hds + hh) * Seq * Dh;

    float lmax = -1e30f;
    for (int j = tid; j < kv; j += 256) {
        const v4f* kr = (const v4f*)(Kb + (size_t)j * Dh);
        const v4f* q4 = (const v4f*)q;
        float s = 0.f;
        #pragma unroll
        for (int i = 0; i < Dh / 4; ++i) {
            v4f kk = kr[i], qq = q4[i];
            s += qq.x * kk.x + qq.y * kk.y + qq.z * kk.z + qq.w * kk.w;
        }
        s *= scale;
        sc[j] = s;
        lmax = fmaxf(lmax, s);
    }
    red[tid] = lmax; __syncthreads();
    for (int off = 128; off > 0; off >>= 1) {
        if (tid < off) red[tid] = fmaxf(red[tid], red[tid + off]);
        __syncthreads();
    }
    float m = red[0];
    __syncthreads();

    float lsum = 0.f;
    for (int j = tid; j < kv; j += 256) { float e = __expf(sc[j] - m); sc[j] = e; lsum += e; }
    red[tid] = lsum; __syncthreads();
    for (int off = 128; off > 0; off >>= 1) {
        if (tid < off) red[tid] += red[tid + off];
        __syncthreads();
    }
    float inv = 1.0f / red[0];
    __syncthreads();

    // 256 threads = 64 d-lanes x 4 j-chunks (coalesced V reads across d)
    int d = tid & (Dh - 1);
    int chunk = tid >> 6;
    float acc = 0.f;
    for (int j = chunk; j < kv; j += 4)
        acc += sc[j] * Vb[(size_t)j * Dh + d];
    red[tid] = acc; __syncthreads();
    if (tid < Dh) {
        float o = (red[tid] + red[tid + 64] + red[tid + 128] + red[tid + 192]) * inv;
        O[(size_t)b * Emb + hh * Dh + tid] = o;
    }
}

// ---------------------------------------------------------------------------
// Logits: out[b][v] = dot(hn[b], wte[v]).  Thread-per-v, hn broadcast from LDS,
// wte streamed once (nontemporal vector loads).
// ---------------------------------------------------------------------------
__global__ void logits_kernel(const float* __restrict__ xn, const float* __restrict__ wte,
                              float* __restrict__ out) {
    __shared__ float sh[Bsz * Emb];
    int tid = threadIdx.x;
    for (int i = tid; i < Bsz * Emb; i += 256) sh[i] = xn[i];
    __syncthreads();
    int v = blockIdx.x * 256 + tid;
    if (v >= Voc) return;
    float acc[Bsz] = {};
    const v4f* wr = (const v4f*)(wte + (size_t)v * Emb);
    for (int e4 = 0; e4 < Emb / 4; ++e4) {
        v4f w = __builtin_nontemporal_load(&wr[e4]);
        int e = e4 * 4;
        #pragma unroll
        for (int b = 0; b < Bsz; ++b) {
            const float* hb = sh + b * Emb + e;
            acc[b] += hb[0] * w.x + hb[1] * w.y + hb[2] * w.z + hb[3] * w.w;
        }
    }
    #pragma unroll
    for (int b = 0; b < Bsz; ++b) out[(size_t)b * Voc + v] = acc[b];
}

// ---------------------------------------------------------------------------
extern "C" void kernel_launch(void* const* d_in, const int* in_sizes, int n_in,
                              void* d_out, int out_size, void* d_ws, size_t ws_size,
                              hipStream_t stream) {
    const int*   ids     = (const int*)  d_in[0];
    const int*   posp    = (const int*)  d_in[1];
    const float* past_k  = (const float*)d_in[2];
    const float* past_v  = (const float*)d_in[3];
    const float* wte     = (const float*)d_in[4];
    const float* wpe     = (const float*)d_in[5];
    const float* ln1g    = (const float*)d_in[6];
    const float* ln1b    = (const float*)d_in[7];
    const float* attn_w  = (const float*)d_in[8];
    const float* attn_b  = (const float*)d_in[9];
    const float* proj_w  = (const float*)d_in[10];
    const float* proj_b  = (const float*)d_in[11];
    const float* ln2g    = (const float*)d_in[12];
    const float* ln2b    = (const float*)d_in[13];
    const float* fc_w    = (const float*)d_in[14];
    const float* fc_b    = (const float*)d_in[15];
    const float* fcp_w   = (const float*)d_in[16];
    const float* fcp_b   = (const float*)d_in[17];
    const float* lnfg    = (const float*)d_in[18];
    const float* lnfb    = (const float*)d_in[19];

    float* out    = (float*)d_out;
    float* logits = out;
    float* nk     = out + (size_t)Bsz * Voc;            // new_keys
    float* nv     = nk + Lyr * BHSD;                    // new_values

    float* ws    = (float*)d_ws;
    float* h     = ws;                 // 8 x 768
    float* xn    = ws + 6144;          // 8 x 768
    float* qkv   = ws + 12288;         // 8 x 2304
    float* ao    = ws + 30720;         // 8 x 768
    float* mid   = ws + 36864;         // 8 x 3072
    float* parts = ws + 61440;         // up to 4 x (8 x 3072) = 16 x (8 x 768) = 98304 floats

    // Pass-through of the KV caches into the output tuple (bulk of traffic).
    (void)hipMemcpyAsync(nk, past_k, Lyr * BHSD * sizeof(float), hipMemcpyDeviceToDevice, stream);
    (void)hipMemcpyAsync(nv, past_v, Lyr * BHSD * sizeof(float), hipMemcpyDeviceToDevice, stream);

    embed_kernel<<<(Bsz * Emb + 255) / 256, 256, 0, stream>>>(ids, posp, wte, wpe, h);

    for (int l = 0; l < Lyr; ++l) {
        layernorm_kernel<<<Bsz, 256, 0, stream>>>(h, ln1g + l * Emb, ln1b + l * Emb, xn);

        // qkv = xn @ attn_w[l] + attn_b[l]              (K=768, N=2304, SK=4)
        gemm_splitk_wmma<<<dim3(3 * Emb / 128, Emb / KCH), 256, 0, stream>>>(
            xn, attn_w + (size_t)l * Emb * 3 * Emb, parts, Emb, 3 * Emb);
        reduce_epilogue_kernel<<<(Bsz * 3 * Emb + 255) / 256, 256, 0, stream>>>(
            parts, attn_b + (size_t)l * 3 * Emb, nullptr, qkv,
            Bsz * 3 * Emb, 3 * Emb, Emb / KCH, 0);

        scatter_kv_kernel<<<(Bsz * Hds * Dh + 255) / 256, 256, 0, stream>>>(
            qkv, nk + (size_t)l * BHSD, nv + (size_t)l * BHSD, posp);

        attention_kernel<<<Bsz * Hds, 256, 0, stream>>>(
            qkv, nk + (size_t)l * BHSD, nv + (size_t)l * BHSD, posp, ao);

        // h = h + ao @ proj_w[l] + proj_b[l]            (K=768, N=768, SK=4)
        gemm_splitk_wmma<<<dim3(Emb / 128, Emb / KCH), 256, 0, stream>>>(
            ao, proj_w + (size_t)l * Emb * Emb, parts, Emb, Emb);
        reduce_epilogue_kernel<<<(Bsz * Emb + 255) / 256, 256, 0, stream>>>(
            parts, proj_b + (size_t)l * Emb, h, h,
            Bsz * Emb, Emb, Emb / KCH, 0);

        layernorm_kernel<<<Bsz, 256, 0, stream>>>(h, ln2g + l * Emb, ln2b + l * Emb, xn);

        // mid = gelu(xn @ fc_w[l] + fc_b[l])            (K=768, N=3072, SK=4)
        gemm_splitk_wmma<<<dim3(4 * Emb / 128, Emb / KCH), 256, 0, stream>>>(
            xn, fc_w + (size_t)l * Emb * 4 * Emb, parts, Emb, 4 * Emb);
        reduce_epilogue_kernel<<<(Bsz * 4 * Emb + 255) / 256, 256, 0, stream>>>(
            parts, fc_b + (size_t)l * 4 * Emb, nullptr, mid,
            Bsz * 4 * Emb, 4 * Emb, Emb / KCH, 1);

        // h = h + mid @ fcp_w[l] + fcp_b[l]             (K=3072, N=768, SK=16)
        gemm_splitk_wmma<<<dim3(Emb / 128, 4 * Emb / KCH), 256, 0, stream>>>(
            mid, fcp_w + (size_t)l * 4 * Emb * Emb, parts, 4 * Emb, Emb);
        reduce_epilogue_kernel<<<(Bsz * Emb + 255) / 256, 256, 0, stream>>>(
            parts, fcp_b + (size_t)l * Emb, h, h,
            Bsz * Emb, Emb, 4 * Emb / KCH, 0);
    }

    layernorm_kernel<<<Bsz, 256, 0, stream>>>(h, lnfg, lnfb, xn);
    logits_kernel<<<(Voc + 255) / 256, 256, 0, stream>>>(xn, wte, logits);
}